// ASDSSMWrapper_84026740179567
// MI455X (gfx1250) — compile-verified
//
#include <hip/hip_runtime.h>

// Problem constants (from reference): N=128, L=512, C=64, d_inner=128
#define T_TOKENS 65536          // N*L
#define C_DIM    64
#define D_INNER  128
#define XZ_COLS  256
#define D_STATE  8
#define DT_RANK  4
#define CHUNK    256
#define NCHUNK   (T_TOKENS / CHUNK)   // 256

typedef float v2f __attribute__((ext_vector_type(2)));
typedef float v8f __attribute__((ext_vector_type(8)));

__device__ __forceinline__ float silu_f(float x) { return x / (1.0f + expf(-x)); }
__device__ __forceinline__ float softplus_f(float x) {
    // numerically stable log(1+exp(x)) = max(x,0) + log1p(exp(-|x|))
    return fmaxf(x, 0.0f) + log1pf(expf(-fabsf(x)));
}

// ---------------------------------------------------------------------------
// Kernel 1: hidden = x + x_res  (written into d_out second half; it is the
// second tuple output AND the input to GEMM1 / residual of GEMM3)
// ---------------------------------------------------------------------------
__global__ __launch_bounds__(256) void k_hidden_add(const float* __restrict__ x,
                                                    const float* __restrict__ xr,
                                                    float* __restrict__ hidden) {
    int i = blockIdx.x * blockDim.x + threadIdx.x;   // float4 index
    float4 a = ((const float4*)x)[i];
    float4 b = ((const float4*)xr)[i];
    float4 h;
    h.x = a.x + b.x; h.y = a.y + b.y; h.z = a.z + b.z; h.w = a.w + b.w;
    ((float4*)hidden)[i] = h;
}

// ---------------------------------------------------------------------------
// Kernel 2: GEMM1  xz = hidden(T x 64) @ W_in(64 x 256), fused SiLU.
// blockIdx.y selects column half: half 0 -> xi = silu(xz[:, :128]),
//                                 half 1 -> zs = silu(xz[:, 128:]).
// One wave computes a 16x128 strip with 8 f32 WMMA accumulators.
// W_in half kept transposed in LDS, stride 66 (even -> 8B aligned float2,
// conflict-free: 66 mod 64 = 2).
// ---------------------------------------------------------------------------
__global__ __launch_bounds__(256) void k_gemm1(const float* __restrict__ hidden,
                                               const float* __restrict__ W_in,
                                               float* __restrict__ xi,
                                               float* __restrict__ zs) {
    __shared__ float Bt[128 * 66];
    const int n0 = blockIdx.y * 128;
    for (int idx = threadIdx.x; idx < 128 * 64; idx += 256) {
        int n = idx & 127;           // coalesced along n in global
        int k = idx >> 7;
        Bt[n * 66 + k] = W_in[k * 256 + n0 + n];
    }
    __syncthreads();

    const int wave = threadIdx.x >> 5;
    const int lane = threadIdx.x & 31;
    const int ln   = lane & 15;
    const int hi   = lane >> 4;               // 0: K={0,1}, 1: K={2,3}
    const int m0   = (blockIdx.x * 8 + wave) * 16;

    const float* aPtr = hidden + (size_t)(m0 + ln) * 64 + hi * 2;

    v8f acc[8];
#pragma unroll
    for (int j = 0; j < 8; ++j) acc[j] = (v8f){0,0,0,0,0,0,0,0};

    for (int ks = 0; ks < 16; ++ks) {         // K = 64, 4 per WMMA
        const int kk = ks * 4;
        float2 at = *(const float2*)(aPtr + kk);
        v2f a; a.x = at.x; a.y = at.y;
#pragma unroll
        for (int j = 0; j < 8; ++j) {
            float2 bt = *(const float2*)(&Bt[(16 * j + ln) * 66 + kk + hi * 2]);
            v2f b; b.x = bt.x; b.y = bt.y;
            acc[j] = __builtin_amdgcn_wmma_f32_16x16x4_f32(
                false, a, false, b, (short)0, acc[j], false, false);
        }
    }

    float* dst = (blockIdx.y == 0) ? xi : zs;
#pragma unroll
    for (int j = 0; j < 8; ++j) {
#pragma unroll
        for (int r = 0; r < 8; ++r) {
            int row = m0 + r + hi * 8;
            int col = 16 * j + ln;
            dst[(size_t)row * 128 + col] = silu_f(acc[j][r]);
        }
    }
}

// ---------------------------------------------------------------------------
// Kernel 3: per-token small projections. One wave per token.
//   xdb[20] = xi_row(128) . W_x   (wave reduction, butterfly shfl)
//   dt[128] = softplus(dt_r(4) @ W_dt + b_dt)
//   Bm = xdb[4:12], Cm = xdb[12:20]
// ---------------------------------------------------------------------------
__global__ __launch_bounds__(256) void k_xdbdt(const float* __restrict__ xi,
                                               const float* __restrict__ W_x,
                                               const float* __restrict__ W_dt,
                                               const float* __restrict__ b_dt,
                                               float* __restrict__ dt,
                                               float* __restrict__ Bm,
                                               float* __restrict__ Cm) {
    __shared__ float sWx[128 * 20];
    __shared__ float sWdt[4 * 128];
    __shared__ float sB[128];
    for (int i = threadIdx.x; i < 128 * 20; i += 256) sWx[i] = W_x[i];
    for (int i = threadIdx.x; i < 512; i += 256) sWdt[i] = W_dt[i];
    if (threadIdx.x < 128) sB[threadIdx.x] = b_dt[threadIdx.x];
    __syncthreads();

    const int wave = threadIdx.x >> 5;
    const int lane = threadIdx.x & 31;
    const int t    = blockIdx.x * 8 + wave;

    float4 x4 = *(const float4*)(xi + (size_t)t * 128 + lane * 4);
    const int k0 = lane * 4;

    float sums[20];
#pragma unroll
    for (int j = 0; j < 20; ++j) {
        float p = x4.x * sWx[(k0 + 0) * 20 + j] + x4.y * sWx[(k0 + 1) * 20 + j]
                + x4.z * sWx[(k0 + 2) * 20 + j] + x4.w * sWx[(k0 + 3) * 20 + j];
        p += __shfl_xor(p, 16);
        p += __shfl_xor(p, 8);
        p += __shfl_xor(p, 4);
        p += __shfl_xor(p, 2);
        p += __shfl_xor(p, 1);
        sums[j] = p;                  // every lane holds the full sum
    }
    if (lane == 0) {
        *(float4*)(Bm + (size_t)t * 8)     = make_float4(sums[4],  sums[5],  sums[6],  sums[7]);
        *(float4*)(Bm + (size_t)t * 8 + 4) = make_float4(sums[8],  sums[9],  sums[10], sums[11]);
        *(float4*)(Cm + (size_t)t * 8)     = make_float4(sums[12], sums[13], sums[14], sums[15]);
        *(float4*)(Cm + (size_t)t * 8 + 4) = make_float4(sums[16], sums[17], sums[18], sums[19]);
    }
    float4 o;
    float* op = (float*)&o;
#pragma unroll
    for (int i = 0; i < 4; ++i) {
        int d = lane * 4 + i;
        float w = sums[0] * sWdt[d]       + sums[1] * sWdt[128 + d]
                + sums[2] * sWdt[256 + d] + sums[3] * sWdt[384 + d] + sB[d];
        op[i] = softplus_f(w);
    }
    *(float4*)(dt + (size_t)t * 128 + lane * 4) = o;
}

// ---------------------------------------------------------------------------
// Chunked parallel scan over T (reference scans the flattened N*L axis).
// Thread (d,s): d = channel (0..127), s = state (0..7); s fastest in lane.
// Pass 1: per chunk, compute aggregate (P = prod a, hB = local scan end).
// ---------------------------------------------------------------------------
__global__ __launch_bounds__(256) void k_scan1(const float* __restrict__ dt,
                                               const float* __restrict__ xi,
                                               const float* __restrict__ Bm,
                                               const float* __restrict__ A_log,
                                               float* __restrict__ chunkA,
                                               float* __restrict__ chunkB) {
    const int c = blockIdx.x >> 2;
    const int d = ((blockIdx.x & 3) * 32) + (threadIdx.x >> 3);
    const int s = threadIdx.x & 7;
    const float Asd = -expf(A_log[d * 8 + s]);

    const float* dtp = dt + (size_t)c * CHUNK * 128 + d;
    const float* xip = xi + (size_t)c * CHUNK * 128 + d;
    const float* bmp = Bm + (size_t)c * CHUNK * 8 + s;

    float h = 0.0f, P = 1.0f;
    for (int i = 0; i < CHUNK; ++i) {
        float dtv = dtp[i * 128];
        float a   = expf(dtv * Asd);
        float b   = dtv * xip[i * 128] * bmp[i * 8];
        h = fmaf(a, h, b);
        P *= a;
    }
    chunkA[c * 1024 + d * 8 + s] = P;
    chunkB[c * 1024 + d * 8 + s] = h;
}

// Pass 2: sequential combine across the 256 chunk aggregates (1024 channels).
__global__ __launch_bounds__(1024) void k_scan2(const float* __restrict__ chunkA,
                                                const float* __restrict__ chunkB,
                                                float* __restrict__ hInit) {
    const int ch = threadIdx.x;      // 0..1023 == d*8+s
    float h = 0.0f;
    for (int c = 0; c < NCHUNK; ++c) {
        hInit[c * 1024 + ch] = h;
        h = fmaf(chunkA[c * 1024 + ch], h, chunkB[c * 1024 + ch]);
    }
}

// Pass 3: replay with correct initial state, fuse y = (sum_s h*C) + D*xi,
// gate with silu(z). s-reduction via 8-lane shfl_xor butterfly.
__global__ __launch_bounds__(256) void k_scan3(const float* __restrict__ dt,
                                               const float* __restrict__ xi,
                                               const float* __restrict__ Bm,
                                               const float* __restrict__ Cm,
                                               const float* __restrict__ A_log,
                                               const float* __restrict__ Dvec,
                                               const float* __restrict__ zs,
                                               const float* __restrict__ hInit,
                                               float* __restrict__ y) {
    const int c = blockIdx.x >> 2;
    const int d = ((blockIdx.x & 3) * 32) + (threadIdx.x >> 3);
    const int s = threadIdx.x & 7;
    const float Asd = -expf(A_log[d * 8 + s]);
    const float Dd  = Dvec[d];

    const float* dtp = dt + (size_t)c * CHUNK * 128 + d;
    const float* xip = xi + (size_t)c * CHUNK * 128 + d;
    const float* zsp = zs + (size_t)c * CHUNK * 128 + d;
    const float* bmp = Bm + (size_t)c * CHUNK * 8 + s;
    const float* cmp = Cm + (size_t)c * CHUNK * 8 + s;
    float* yp = y + (size_t)c * CHUNK * 128 + d;

    float h = hInit[c * 1024 + d * 8 + s];
    for (int i = 0; i < CHUNK; ++i) {
        float dtv = dtp[i * 128];
        float xiv = xip[i * 128];
        float a   = expf(dtv * Asd);
        h = fmaf(a, h, dtv * xiv * bmp[i * 8]);
        float p = h * cmp[i * 8];
        p += __shfl_xor(p, 1);
        p += __shfl_xor(p, 2);
        p += __shfl_xor(p, 4);
        if (s == 0) {
            yp[i * 128] = (p + Dd * xiv) * zsp[i * 128];
        }
    }
}

// ---------------------------------------------------------------------------
// Kernel 7: GEMM3  out = y(T x 128) @ W_out(128 x 64) + hidden  (x_out).
// One wave: 16x64 strip, 4 f32 WMMA accumulators, K=128 in 32 steps.
// ---------------------------------------------------------------------------
__global__ __launch_bounds__(256) void k_gemm3(const float* __restrict__ y,
                                               const float* __restrict__ W_out,
                                               const float* __restrict__ hidden,
                                               float* __restrict__ xout) {
    __shared__ float Wt[64 * 130];   // transposed, stride 130 (even, conflict-free)
    for (int idx = threadIdx.x; idx < 64 * 128; idx += 256) {
        int n = idx & 63;
        int k = idx >> 6;
        Wt[n * 130 + k] = W_out[k * 64 + n];
    }
    __syncthreads();

    const int wave = threadIdx.x >> 5;
    const int lane = threadIdx.x & 31;
    const int ln   = lane & 15;
    const int hi   = lane >> 4;
    const int m0   = (blockIdx.x * 8 + wave) * 16;

    const float* aPtr = y + (size_t)(m0 + ln) * 128 + hi * 2;

    v8f acc[4];
#pragma unroll
    for (int j = 0; j < 4; ++j) acc[j] = (v8f){0,0,0,0,0,0,0,0};

    for (int ks = 0; ks < 32; ++ks) {
        const int kk = ks * 4;
        float2 at = *(const float2*)(aPtr + kk);
        v2f a; a.x = at.x; a.y = at.y;
#pragma unroll
        for (int j = 0; j < 4; ++j) {
            float2 bt = *(const float2*)(&Wt[(16 * j + ln) * 130 + kk + hi * 2]);
            v2f b; b.x = bt.x; b.y = bt.y;
            acc[j] = __builtin_amdgcn_wmma_f32_16x16x4_f32(
                false, a, false, b, (short)0, acc[j], false, false);
        }
    }

#pragma unroll
    for (int j = 0; j < 4; ++j) {
#pragma unroll
        for (int r = 0; r < 8; ++r) {
            int row = m0 + r + hi * 8;
            int col = 16 * j + ln;
            size_t off = (size_t)row * 64 + col;
            xout[off] = acc[j][r] + hidden[off];
        }
    }
}

// ---------------------------------------------------------------------------
// Launch: inputs (dict order): 0 x, 1 x_res, 2 scale_id, 3 W_in, 4 W_x,
// 5 W_dt, 6 b_dt, 7 A_log, 8 D, 9 W_out. Output: [x_out | hidden] flat.
// ---------------------------------------------------------------------------
extern "C" void kernel_launch(void* const* d_in, const int* in_sizes, int n_in,
                              void* d_out, int out_size, void* d_ws, size_t ws_size,
                              hipStream_t stream) {
    const float* x     = (const float*)d_in[0];
    const float* xr    = (const float*)d_in[1];
    const float* W_in  = (const float*)d_in[3];
    const float* W_x   = (const float*)d_in[4];
    const float* W_dt  = (const float*)d_in[5];
    const float* b_dt  = (const float*)d_in[6];
    const float* A_log = (const float*)d_in[7];
    const float* Dv    = (const float*)d_in[8];
    const float* W_out = (const float*)d_in[9];

    float* out    = (float*)d_out;
    float* hidden = out + (size_t)T_TOKENS * C_DIM;   // second tuple output

    float* w  = (float*)d_ws;
    float* xi = w;                                    // T*128
    float* zs = xi + (size_t)T_TOKENS * 128;          // T*128
    float* dt = zs + (size_t)T_TOKENS * 128;          // T*128
    float* yb = dt + (size_t)T_TOKENS * 128;          // T*128
    float* Bm = yb + (size_t)T_TOKENS * 128;          // T*8
    float* Cm = Bm + (size_t)T_TOKENS * 8;            // T*8
    float* cA = Cm + (size_t)T_TOKENS * 8;            // NCHUNK*1024
    float* cB = cA + NCHUNK * 1024;                   // NCHUNK*1024
    float* hI = cB + NCHUNK * 1024;                   // NCHUNK*1024

    k_hidden_add<<<(T_TOKENS * C_DIM / 4) / 256, 256, 0, stream>>>(x, xr, hidden);
    k_gemm1<<<dim3(T_TOKENS / 16 / 8, 2), 256, 0, stream>>>(hidden, W_in, xi, zs);
    k_xdbdt<<<T_TOKENS / 8, 256, 0, stream>>>(xi, W_x, W_dt, b_dt, dt, Bm, Cm);
    k_scan1<<<NCHUNK * 4, 256, 0, stream>>>(dt, xi, Bm, A_log, cA, cB);
    k_scan2<<<1, 1024, 0, stream>>>(cA, cB, hI);
    k_scan3<<<NCHUNK * 4, 256, 0, stream>>>(dt, xi, Bm, Cm, A_log, Dv, zs, hI, yb);
    k_gemm3<<<T_TOKENS / 16 / 8, 256, 0, stream>>>(yb, W_out, hidden, out);
}